// FuseSliceModule_25314537242671
// MI455X (gfx1250) — compile-verified
//
#include <hip/hip_runtime.h>
#include <hip/hip_bf16.h>

// Problem constants (match reference exactly)
#define ROWS_TOTAL  (2048 * 64)   // 131072 flattened rows
#define FEAT        512           // columns per row
#define SLICE_LEN   64            // floats copied per row per slice
#define NUM_SLICES  16
#define TILE_ROWS   128           // rows per TDM tile (32 KB LDS)
#define CHUNKS      (ROWS_TOTAL / TILE_ROWS)   // 1024 chunks per slice

typedef __attribute__((ext_vector_type(4))) unsigned int u32x4;
typedef __attribute__((ext_vector_type(8))) int          i32x8;
typedef __attribute__((ext_vector_type(4))) int          i32x4;

#if defined(__HIP_DEVICE_COMPILE__) && \
    __has_builtin(__builtin_amdgcn_tensor_load_to_lds) && \
    __has_builtin(__builtin_amdgcn_tensor_store_from_lds) && \
    __has_builtin(__builtin_amdgcn_s_wait_tensorcnt)
#define USE_TDM 1
#else
#define USE_TDM 0
#endif

// One wave per block. Each block moves one tile:
//   slice s, rows [row0, row0+TILE_ROWS) of the flattened (131072 x 512) input,
//   columns [col, col+64) where col = slices_index[s],
// into out + (s*ROWS_TOTAL + row0)*64 (fully contiguous destination).
__global__ __launch_bounds__(32)
void FuseSlice_tdm_kernel(const float* __restrict__ in,
                          const int*   __restrict__ sl,
                          float*       __restrict__ out) {
    __shared__ float tile[TILE_ROWS * SLICE_LEN];   // 32 KB

    const unsigned b     = blockIdx.x;
    const unsigned s     = b >> 10;          // CHUNKS == 1024
    const unsigned chunk = b & 1023u;
    const unsigned row0  = chunk * TILE_ROWS;
    const int      col   = sl[s];            // uniform -> scalar load

    const unsigned long long ga_in =
        (unsigned long long)(size_t)in +
        ((unsigned long long)row0 * FEAT + (unsigned)col) * 4ull;
    const unsigned long long ga_out =
        (unsigned long long)(size_t)out +
        ((unsigned long long)s * ROWS_TOTAL + row0) * (SLICE_LEN * 4ull);

#if USE_TDM
    const unsigned lds = (unsigned)(size_t)(&tile[0]);  // LDS byte offset

    // ---- Load descriptor: 2D tile, 64 cols x 128 rows, row stride 512 elems
    u32x4 g0l;
    g0l.x = 1u;                                   // count=1 (valid user D#)
    g0l.y = lds;                                  // lds_addr
    g0l.z = (unsigned)(ga_in & 0xFFFFFFFFu);      // global_addr[31:0]
    g0l.w = (unsigned)(ga_in >> 32) | 0x80000000u;// global_addr[56:32] | type=2

    i32x8 g1l;
    g1l[0] = (2 << 16);                           // data_size = 4 bytes
    g1l[1] = (int)((FEAT & 0xFFFFu) << 16);       // tensor_dim0[15:0] = 512
    g1l[2] = (int)((FEAT >> 16) |
                   ((ROWS_TOTAL & 0xFFFFu) << 16));          // dim0 hi | dim1 lo
    g1l[3] = (int)(((unsigned)ROWS_TOTAL >> 16) |
                   ((unsigned)SLICE_LEN << 16));             // dim1 hi | tile_dim0=64
    g1l[4] = TILE_ROWS;                           // tile_dim1=128, tile_dim2=0
    g1l[5] = FEAT;                                // tensor_dim0_stride[31:0] = 512
    g1l[6] = 0;                                   // stride0 hi | stride1 lo
    g1l[7] = 0;

    // ---- Store descriptor: 1D contiguous tile of 8192 elements
    const unsigned SN = TILE_ROWS * SLICE_LEN;    // 8192
    u32x4 g0s;
    g0s.x = 1u;
    g0s.y = lds;
    g0s.z = (unsigned)(ga_out & 0xFFFFFFFFu);
    g0s.w = (unsigned)(ga_out >> 32) | 0x80000000u;

    i32x8 g1s;
    g1s[0] = (2 << 16);                           // data_size = 4 bytes
    g1s[1] = (int)((SN & 0xFFFFu) << 16);         // tensor_dim0[15:0]
    g1s[2] = (int)((SN >> 16) | (1u << 16));      // dim0 hi | tensor_dim1 = 1
    g1s[3] = (int)(SN << 16);                     // dim1 hi=0 | tile_dim0 = 8192
    g1s[4] = 1;                                   // tile_dim1 = 1, tile_dim2 = 0
    g1s[5] = (int)SN;                             // tensor_dim0_stride = 8192
    g1s[6] = 0;
    g1s[7] = 0;

    i32x4 gz4 = {0, 0, 0, 0};                     // groups 2/3: dims unused
    i32x8 gz8 = {0, 0, 0, 0, 0, 0, 0, 0};         // extra arg (clang-23 6-arg form)

    __builtin_amdgcn_tensor_load_to_lds(g0l, g1l, gz4, gz4, gz8, 0);
    __builtin_amdgcn_s_wait_tensorcnt(0);         // tile resident in LDS
    __builtin_amdgcn_tensor_store_from_lds(g0s, g1s, gz4, gz4, gz8, 0);
    __builtin_amdgcn_s_wait_tensorcnt(0);         // drained before endpgm
#else
    // Fallback (toolchain without TDM builtins): scalar b32 copy.
    // Column offset is only 4-byte aligned, so no float4 on the source.
    const float* src = (const float*)(size_t)ga_in;
    float*       dst = (float*)(size_t)ga_out;
    for (unsigned t = threadIdx.x; t < TILE_ROWS * SLICE_LEN; t += 32) {
        const unsigned r = t / SLICE_LEN;
        const unsigned j = t % SLICE_LEN;
        dst[r * SLICE_LEN + j] = src[(size_t)r * FEAT + j];
    }
    (void)tile;
#endif
}

extern "C" void kernel_launch(void* const* d_in, const int* in_sizes, int n_in,
                              void* d_out, int out_size, void* d_ws, size_t ws_size,
                              hipStream_t stream) {
    const float* in  = (const float*)d_in[0];
    const int*   sl  = (const int*)d_in[1];
    float*       out = (float*)d_out;
    (void)in_sizes; (void)n_in; (void)out_size; (void)d_ws; (void)ws_size;

    dim3 grid(NUM_SLICES * CHUNKS);   // 16 * 1024 = 16384 blocks
    dim3 block(32);                   // one wave per block (TDM is wave-level)
    FuseSlice_tdm_kernel<<<grid, block, 0, stream>>>(in, sl, out);
}